// proxy_net_26783416058451
// MI455X (gfx1250) — compile-verified
//
#include <hip/hip_runtime.h>
#include <cstddef>
#include <cstdint>

// ---------------- problem constants (baked from the reference) ----------------
#define T_STEPS   2500
#define TT        100                       // time-tile (2500 = 25 * 100)
#define NTILES    (T_STEPS / TT)            // 25
#define ROWS      32                        // neurons per block
#define NTHR      64                        // wave0: daq scan, wave1: LIF scan
#define TILE_F4   (ROWS * TT / 4)           // 800 float4 per tile
#define FULL_ITERS (TILE_F4 / NTHR)         // 12 guard-free b128 issues / thread
#define TAIL_F4   (TILE_F4 - FULL_ITERS * NTHR)  // 32 remaining float4 (tid < 32)
#define MIN_LOADS FULL_ITERS                // 12 = min per-wave issue count

#define A_SYN  0.60653065971263342f         // exp(-dt/0.0002) = exp(-0.5)
#define B_MEM  0.99004983374916811f         // exp(-dt/0.01)   = exp(-0.01)
#define EL     (-70.0f)
#define V_TH   (-50.0f)
#define V_RST  (-70.0f)

// Pointer types matching the gfx1250 async-to-LDS builtin signature:
//   b128 variant takes (v4i addrspace(1)*, v4i addrspace(3)*, imm offset, imm cpol)
typedef int v4i __attribute__((ext_vector_type(4)));
typedef __attribute__((address_space(1))) v4i* gv4_t;   // global
typedef __attribute__((address_space(3))) v4i* lv4_t;   // LDS

__device__ __forceinline__ void async_copy_b128(const void* g, void* l) {
  // gfx1250: GLOBAL_LOAD_ASYNC_TO_LDS_B128 (per-lane global src, per-lane LDS dst)
  __builtin_amdgcn_global_load_async_to_lds_b128((gv4_t)g, (lv4_t)l, 0, 0);
}

__device__ __forceinline__ float fast_tanh(float v) {
  // gfx1250-only transcendental; v_nop covers the TRANS RAW hazard window
  float r;
  asm("v_tanh_f32 %0, %1\n\tv_nop" : "=v"(r) : "v"(v));
  return r;
}

__global__ __launch_bounds__(NTHR, 1) void lif_dual_scan(
    const float* __restrict__ x,        // [B, T] row-major
    const float* __restrict__ scale_p,  // scalar
    const float* __restrict__ thr_p,    // scalar
    float* __restrict__ out,            // [4, T, B] concatenated
    int Bn)
{
  __shared__ __align__(16) float tile[2][ROWS * TT];   // 2 x 12.8 KB double buffer

  const int tid = threadIdx.x;
  const int b0  = blockIdx.x * ROWS;
  const float scale = scale_p[0];
  const float thr   = thr_p[0];

  const size_t TB = (size_t)T_STEPS * (size_t)Bn;
  float* mem_daq = out;            // output 0: mem_daq   [T,B]
  float* out_mem = out + TB;       // output 1: out_mem   [T,B]
  float* spk_daq = out + 2 * TB;   // output 2: spikes_daq[T,B]
  float* spk_lif = out + 3 * TB;   // output 3: spk_      [T,B]

  // Cooperative async stage of tile k (ROWS x TT floats) into buffer `sel`.
  // Global reads are 16B-coalesced along t; LDS layout [row][tt], stride 100.
  // 12 guard-free issues + one wave0-only tail issue => no per-issue exec juggling.
  auto issue_tile = [&](int k, int sel) {
    const float* gbase = x + (size_t)b0 * T_STEPS + (size_t)k * TT;
    float* lbase = &tile[sel][0];
#pragma unroll
    for (int c = 0; c < FULL_ITERS; ++c) {
      int f4  = c * NTHR + tid;               // 0..767 < 800: always in range
      int row = f4 / (TT / 4);
      int t4  = f4 % (TT / 4);
      async_copy_b128(gbase + (size_t)row * T_STEPS + t4 * 4,
                      lbase + row * TT + t4 * 4);
    }
    if (tid < TAIL_F4) {                      // whole-wave skip for wave1
      int f4  = FULL_ITERS * NTHR + tid;      // 768..799
      int row = f4 / (TT / 4);
      int t4  = f4 % (TT / 4);
      async_copy_b128(gbase + (size_t)row * T_STEPS + t4 * 4,
                      lbase + row * TT + t4 * 4);
    }
  };

  issue_tile(0, 0);

  const bool is_daq = (tid < ROWS);          // wave0 = daq, wave1 = LIF
  const int  r  = is_daq ? tid : (tid - ROWS);
  const int  bg = b0 + r;

  float syn = 0.0f, mem = 0.0f;              // daq state
  float V   = EL;                            // LIF state

  for (int k = 0; k < NTILES; ++k) {
    const int sel = k & 1;
    if (k + 1 < NTILES) {
      issue_tile(k + 1, sel ^ 1);            // prefetch next tile (other buffer)
      // in-order completion: <=MIN_LOADS outstanding => tile k fully resident
      __builtin_amdgcn_s_wait_asynccnt(MIN_LOADS);
    } else {
      __builtin_amdgcn_s_wait_asynccnt(0);
    }
    __syncthreads();                         // all waves' tile-k loads visible

    const float* tp = &tile[sel][r * TT];
    const int tbase = k * TT;

    if (is_daq) {
#pragma unroll 4
      for (int tt = 0; tt < TT; ++tt) {
        float cur = tp[tt];
        syn = A_SYN * syn + cur * scale;
        mem = B_MEM * mem + syn;
        float spk = (mem > thr) ? 1.0f : 0.0f;
        mem = fast_tanh(mem - spk * thr);    // reset-by-subtraction then tanh
        size_t o = (size_t)(tbase + tt) * (size_t)Bn + (size_t)bg;
        mem_daq[o] = mem;                    // coalesced: lane index = b
        spk_daq[o] = spk;
      }
    } else {
#pragma unroll 4
      for (int tt = 0; tt < TT; ++tt) {
        float i = tp[tt] * 1000.0f;
        V += 0.01f * ((EL - V) + 0.1f * i);  // dt/tau_m = 0.01
        float s = (V > V_TH) ? 1.0f : 0.0f;
        V = (V > V_TH) ? V_RST : V;
        size_t o = (size_t)(tbase + tt) * (size_t)Bn + (size_t)bg;
        out_mem[o] = (V + 70.0f) * (1.0f / 30.0f);
        spk_lif[o] = s;
      }
    }
    __syncthreads();                         // tile k consumed before tile k+2 overwrites
  }
}

extern "C" void kernel_launch(void* const* d_in, const int* in_sizes, int n_in,
                              void* d_out, int out_size, void* d_ws, size_t ws_size,
                              hipStream_t stream) {
  (void)n_in; (void)out_size; (void)d_ws; (void)ws_size;
  const float* x     = (const float*)d_in[0];
  const float* scale = (const float*)d_in[1];
  const float* thr   = (const float*)d_in[2];
  float* out = (float*)d_out;

  const int Bn = in_sizes[0] / T_STEPS;      // 8192
  const int nblocks = Bn / ROWS;             // 256 blocks x 2 waves = 512 waves (ceiling)
  lif_dual_scan<<<dim3(nblocks), dim3(NTHR), 0, stream>>>(x, scale, thr, out, Bn);
}